// DenormProject_39771397161206
// MI455X (gfx1250) — compile-verified
//
#include <hip/hip_runtime.h>
#include <math.h>

// ---------------- problem constants (match reference) ----------------
#define NVIEW  5
#define BATCH  2
#define C_IMG  3
#define C_FEATC 32
#define H_IMG  512
#define W_IMG  512
#define H_F    128
#define W_F    128
#define NBINS  (80 * 80 * 20)      // 128000
#define IN_DIM 73                  // 3 + 2*(3+32)
#define HID    256

// ---------------- tiling ----------------
#define KPAD    80                 // IN_DIM padded to multiple of 16 (zero-filled)
#define ASTRIDE 84                 // LDS row stride for A tile (odd*4 dwords -> conflict-free)
#define MTILE   128
#define THREADS 256

typedef __attribute__((ext_vector_type(2))) float v2f;
typedef __attribute__((ext_vector_type(8))) float v8f;

struct Corner {
    int   x0, x1, y0, y1;
    float w00, w10, w01, w11;
};

__device__ __forceinline__ Corner make_corner(float gx, float gy, int H, int W) {
    Corner c;
    float ix  = (gx + 1.0f) * 0.5f * (float)(W - 1);
    float iy  = (gy + 1.0f) * 0.5f * (float)(H - 1);
    float x0f = floorf(ix), y0f = floorf(iy);
    float wx1 = ix - x0f,  wx0 = 1.0f - wx1;
    float wy1 = iy - y0f,  wy0 = 1.0f - wy1;
    float x1f = x0f + 1.0f, y1f = y0f + 1.0f;
    float vx0 = (x0f >= 0.0f && x0f <= (float)(W - 1)) ? 1.0f : 0.0f;
    float vx1 = (x1f >= 0.0f && x1f <= (float)(W - 1)) ? 1.0f : 0.0f;
    float vy0 = (y0f >= 0.0f && y0f <= (float)(H - 1)) ? 1.0f : 0.0f;
    float vy1 = (y1f >= 0.0f && y1f <= (float)(H - 1)) ? 1.0f : 0.0f;
    c.x0 = (int)fminf(fmaxf(x0f, 0.0f), (float)(W - 1));
    c.x1 = (int)fminf(fmaxf(x1f, 0.0f), (float)(W - 1));
    c.y0 = (int)fminf(fmaxf(y0f, 0.0f), (float)(H - 1));
    c.y1 = (int)fminf(fmaxf(y1f, 0.0f), (float)(H - 1));
    c.w00 = wx0 * wy0 * vx0 * vy0;
    c.w10 = wx1 * wy0 * vx1 * vy0;
    c.w01 = wx0 * wy1 * vx0 * vy1;
    c.w11 = wx1 * wy1 * vx1 * vy1;
    return c;
}

// Sample channels [c_begin, c_end) of img ([NVIEW, BATCH, C, H, W]) at point
// n_pt using all NVIEW grids; write mean -> Arow[meanOff + c], exp(-var) ->
// Arow[varOff + c].
__device__ __forceinline__ void bilinear_stats(
    const float* __restrict__ img, const float* __restrict__ grid,
    int b, int n_pt, int C, int H, int W, int c_begin, int c_end,
    float* __restrict__ Arow, int meanOff, int varOff)
{
    Corner cr[NVIEW];
#pragma unroll
    for (int v = 0; v < NVIEW; ++v) {
        float gx = grid[((size_t)v * NBINS + n_pt) * 2 + 0];
        float gy = grid[((size_t)v * NBINS + n_pt) * 2 + 1];
        cr[v] = make_corner(gx, gy, H, W);
    }
    for (int c = c_begin; c < c_end; ++c) {
        float s = 0.0f, s2 = 0.0f;
#pragma unroll
        for (int v = 0; v < NVIEW; ++v) {
            const float* p = img + ((((size_t)v * BATCH + b) * C + c) * (size_t)H) * W;
            float val = cr[v].w00 * p[(size_t)cr[v].y0 * W + cr[v].x0]
                      + cr[v].w10 * p[(size_t)cr[v].y0 * W + cr[v].x1]
                      + cr[v].w01 * p[(size_t)cr[v].y1 * W + cr[v].x0]
                      + cr[v].w11 * p[(size_t)cr[v].y1 * W + cr[v].x1];
            s  += val;
            s2 += val * val;
        }
        float mean = s * (1.0f / NVIEW);
        float var  = s2 * (1.0f / NVIEW) - mean * mean;
        Arow[meanOff + c] = mean;
        Arow[varOff  + c] = expf(-var);
    }
}

__global__ void __launch_bounds__(THREADS)
fused_project_mlp_kernel(
    const float* __restrict__ denorm_imgs,  // [NVIEW,B,3,512,512]
    const float* __restrict__ feats2d,      // [NVIEW,B,32,128,128]
    const float* __restrict__ sample_grid,  // [NVIEW,NBINS,2]
    const float* __restrict__ points,       // [NBINS,3]
    const float* __restrict__ W1,           // [73,256]
    const float* __restrict__ b1,           // [256]
    const float* __restrict__ W2,           // [256,1]
    const float* __restrict__ b2,           // [1]
    float* __restrict__ out)                // [B,NBINS]
{
    extern __shared__ float smem[];
    float* Wlds   = smem;                      // KPAD * HID   (80 KB)
    float* Atile  = Wlds  + KPAD * HID;        // MTILE * ASTRIDE (43 KB)
    float* b1s    = Atile + MTILE * ASTRIDE;   // HID
    float* w2s    = b1s   + HID;               // HID
    float* rowsum = w2s   + HID;               // MTILE

    const int tid = threadIdx.x;
    const int n0  = blockIdx.x * MTILE;
    const int b   = blockIdx.y;

    // ---- cooperative load of W1 (zero-padded to KPAD rows), b1, W2 ----
    for (int i = tid; i < KPAD * HID; i += THREADS) {
        int k = i >> 8;            // / HID
        int n = i & (HID - 1);
        Wlds[i] = (k < IN_DIM) ? W1[k * HID + n] : 0.0f;
    }
    b1s[tid] = b1[tid];
    w2s[tid] = W2[tid];
    if (tid < MTILE) rowsum[tid] = 0.0f;

    // ---- gather + mean / exp(-var) phase: 2 threads per row ----
    const int rowL = tid & (MTILE - 1);
    const int half = tid >> 7;
    const int n_pt = n0 + rowL;
    float* Arow = &Atile[rowL * ASTRIDE];

    if (half == 0) {
        Arow[0] = points[n_pt * 3 + 0];
        Arow[1] = points[n_pt * 3 + 1];
        Arow[2] = points[n_pt * 3 + 2];
        for (int k = IN_DIM; k < ASTRIDE; ++k) Arow[k] = 0.0f;  // zero pad cols
        // img channels c=0..2 -> concat idx c: mean at 3+c, var at 38+c
        bilinear_stats(denorm_imgs, sample_grid, b, n_pt, C_IMG, H_IMG, W_IMG,
                       0, C_IMG, Arow, 3, 38);
        // feat channels c=0..15 -> concat idx 3+c: mean at 6+c, var at 41+c
        bilinear_stats(feats2d, sample_grid, b, n_pt, C_FEATC, H_F, W_F,
                       0, 16, Arow, 6, 41);
    } else {
        // feat channels c=16..31
        bilinear_stats(feats2d, sample_grid, b, n_pt, C_FEATC, H_F, W_F,
                       16, C_FEATC, Arow, 6, 41);
    }
    __syncthreads();

    // ---- GEMM1 (fp32 WMMA 16x16x4) + ReLU + fused GEMM2 dot ----
    const int lane  = tid & 31;
    const int wave  = tid >> 5;           // 8 waves
    const int l15   = lane & 15;
    const int khalf = lane >> 4;          // K-half per ISA A/B layout

#pragma unroll 1
    for (int mt = 0; mt < MTILE / 16; ++mt) {
        float acc[8];
#pragma unroll
        for (int j = 0; j < 8; ++j) acc[j] = 0.0f;

#pragma unroll 1
        for (int nt2 = 0; nt2 < 2; ++nt2) {
            const int nt   = wave * 2 + nt2;      // wave owns hidden cols [32w, 32w+32)
            const int ncol = nt * 16 + l15;
            const float bias = b1s[ncol];
            v8f c;
#pragma unroll
            for (int j = 0; j < 8; ++j) c[j] = bias;   // fold b1 into accumulator

            const float* arow = &Atile[(mt * 16 + l15) * ASTRIDE + 2 * khalf];
#pragma unroll
            for (int kb = 0; kb < KPAD; kb += 4) {
                v2f a = *(const v2f*)(arow + kb);               // A[m][kb+2h .. +1]
                v2f bv;
                bv.x = Wlds[(kb + 2 * khalf)     * HID + ncol]; // B[kb+2h  ][n]
                bv.y = Wlds[(kb + 2 * khalf + 1) * HID + ncol]; // B[kb+2h+1][n]
                c = __builtin_amdgcn_wmma_f32_16x16x4_f32(
                        /*neg_a=*/false, a, /*neg_b=*/false, bv,
                        /*c_mod=*/(short)0, c, /*reuse_a=*/false, /*reuse_b=*/false);
            }
            const float w2v = w2s[ncol];
#pragma unroll
            for (int j = 0; j < 8; ++j) acc[j] += fmaxf(c[j], 0.0f) * w2v;
        }

        // reduce 16-lane halves; C layout: VGPR j <-> rows mt*16 + j (+8 upper half)
#pragma unroll
        for (int j = 0; j < 8; ++j) {
            float v = acc[j];
            v += __shfl_xor(v, 1, 32);
            v += __shfl_xor(v, 2, 32);
            v += __shfl_xor(v, 4, 32);
            v += __shfl_xor(v, 8, 32);
            if (l15 == 0) atomicAdd(&rowsum[mt * 16 + j + 8 * khalf], v);
        }
    }
    __syncthreads();

    // ---- softplus -> alpha ----
    if (tid < MTILE) {
        float d  = rowsum[tid] + b2[0];
        float sp = fmaxf(d, 0.0f) + log1pf(expf(-fabsf(d)));   // softplus, overflow-safe
        float alpha = 1.0f - expf(-sp);
        out[(size_t)b * NBINS + n0 + tid] = alpha;
    }
}

extern "C" void kernel_launch(void* const* d_in, const int* in_sizes, int n_in,
                              void* d_out, int out_size, void* d_ws, size_t ws_size,
                              hipStream_t stream) {
    const float* denorm_imgs = (const float*)d_in[0];
    const float* feats2d     = (const float*)d_in[1];
    const float* sample_grid = (const float*)d_in[2];
    const float* points      = (const float*)d_in[3];
    const float* W1          = (const float*)d_in[4];
    const float* b1          = (const float*)d_in[5];
    const float* W2          = (const float*)d_in[6];
    const float* b2          = (const float*)d_in[7];
    float* out = (float*)d_out;

    dim3 grid(NBINS / MTILE, BATCH);   // (1000, 2)
    size_t lds_bytes = (size_t)(KPAD * HID + MTILE * ASTRIDE + HID + HID + MTILE)
                     * sizeof(float);  // ~124.5 KB
    fused_project_mlp_kernel<<<grid, THREADS, lds_bytes, stream>>>(
        denorm_imgs, feats2d, sample_grid, points, W1, b1, W2, b2, out);
}